// EncodeLayer_72035191488655
// MI455X (gfx1250) — compile-verified
//
#include <hip/hip_runtime.h>

// FPS for 16 clouds x 16384 pts -> 4096 samples/cloud.
// One workgroup per cloud; positions staged into LDS via CDNA5 TDM
// (tensor_load_to_lds), per-thread state held in VGPRs, wave32 shuffle argmax.

#define NCLOUDS 16
#define NPTS    16384
#define NSAMP   4096
#define BLOCK   1024
#define NWAVES  (BLOCK / 32)
#define PPT     (NPTS / BLOCK)   // 16 points per thread

typedef __attribute__((ext_vector_type(4))) unsigned int u32x4;
typedef __attribute__((ext_vector_type(8))) int          i32x8;
typedef __attribute__((ext_vector_type(4))) int          i32x4;

__device__ __forceinline__ unsigned long long umax64(unsigned long long a,
                                                     unsigned long long b) {
    return a > b ? a : b;
}

// 64-bit xor-shuffle built from two 32-bit wave32 shuffles (wave32-safe).
__device__ __forceinline__ unsigned long long shfl_xor_u64(unsigned long long x, int m) {
    unsigned lo = (unsigned)x;
    unsigned hi = (unsigned)(x >> 32);
    lo = __shfl_xor(lo, m, 32);
    hi = __shfl_xor(hi, m, 32);
    return ((unsigned long long)hi << 32) | lo;
}

__global__ __launch_bounds__(BLOCK, 1)
void fps_kernel(const float* __restrict__ pos, float* __restrict__ out) {
    // 192 KB point table (AoS xyz) + reduction scratch: fits CDNA5's 320 KB WGP LDS.
    __shared__ float               s_pos[NPTS * 3];
    __shared__ unsigned long long  s_wred[NWAVES];
    __shared__ unsigned long long  s_bcast;

    const int tid  = threadIdx.x;
    const int wid  = tid >> 5;
    const int lane = tid & 31;
    const int b    = blockIdx.x;
    const float* gpos = pos + (size_t)b * NPTS * 3;

    // ---------------- stage positions: global -> LDS ----------------
#if __has_builtin(__builtin_amdgcn_tensor_load_to_lds) && __has_builtin(__builtin_amdgcn_s_wait_tensorcnt)
    if (wid == 0) {
        const unsigned long long ga    = (unsigned long long)gpos;
        const unsigned           ldsa  = (unsigned)(size_t)(void*)&s_pos[0];
        const unsigned           ELEMS = NPTS * 3;          // 49152 fp32 elements

        u32x4 g0;
        g0[0] = 1u;                                          // count=1 (valid D#)
        g0[1] = ldsa;                                        // lds_addr (bytes)
        g0[2] = (unsigned)(ga & 0xffffffffull);              // global_addr[31:0]
        g0[3] = (unsigned)((ga >> 32) & 0x1ffffffull)        // global_addr[56:32]
              | (2u << 30);                                  // type = 2 ("image")

        unsigned w0 = (2u << 16);                            // data_size=2 -> 4 bytes
        unsigned w1 = (ELEMS & 0xffffu) << 16;               // tensor_dim0[15:0]
        unsigned w2 = ((ELEMS >> 16) & 0xffffu)              // tensor_dim0[31:16]
                    | (1u << 16);                            // tensor_dim1 = 1
        unsigned w3 = (ELEMS & 0xffffu) << 16;               // tile_dim0 = 49152 (1-D tile)
        unsigned w5 = ELEMS;                                 // tensor_dim0_stride[31:0]
        i32x8 g1;
        g1[0] = (int)w0; g1[1] = (int)w1; g1[2] = (int)w2; g1[3] = (int)w3;
        g1[4] = 0;       g1[5] = (int)w5; g1[6] = 0;       g1[7] = 0;

        i32x4 g2 = (i32x4)0;
        i32x4 g3 = (i32x4)0;
        i32x8 g4 = (i32x8)0;   // 6-arg toolchain form (clang-23 / therock-10 headers)

        __builtin_amdgcn_tensor_load_to_lds(g0, g1, g2, g3, g4, 0);
        __builtin_amdgcn_s_wait_tensorcnt(0);
    }
    __syncthreads();
#else
    {
        const float4* src = (const float4*)gpos;
        float4*       dst = (float4*)s_pos;
        for (int j = tid; j < (NPTS * 3) / 4; j += BLOCK) dst[j] = src[j];
    }
    __syncthreads();
#endif

    // ---------------- cache this thread's 16 points + mind in VGPRs ----------------
    float px[PPT], py[PPT], pz[PPT], mind[PPT];
#pragma unroll
    for (int k = 0; k < PPT; ++k) {
        const int i = tid + k * BLOCK;                       // strided ownership
        px[k] = s_pos[3 * i + 0];
        py[k] = s_pos[3 * i + 1];
        pz[k] = s_pos[3 * i + 2];
    }

    // First pick is local point 0 (torch_geometric fps, random_start=False).
    float qx = s_pos[0], qy = s_pos[1], qz = s_pos[2];

    // local fused update + argmax key: (dist_bits << 32) | ~idx
    // dist >= 0 so float bits are order-preserving; ~idx makes ties pick smallest idx.
    unsigned long long lkey = 0;
#pragma unroll
    for (int k = 0; k < PPT; ++k) {
        const float dx = px[k] - qx, dy = py[k] - qy, dz = pz[k] - qz;
        const float d  = dx * dx + dy * dy + dz * dz;
        mind[k] = d;
        const unsigned idx = (unsigned)(tid + k * BLOCK);
        lkey = umax64(lkey, ((unsigned long long)__float_as_uint(d) << 32) | (~idx));
    }

    if (tid == 0) out[(size_t)b * NSAMP] = (float)(b * NPTS);

    // ---------------- 4095 sequential FPS rounds ----------------
    for (int t = 1; t < NSAMP; ++t) {
        // wave32 argmax tree on the packed key
        unsigned long long v = lkey;
#pragma unroll
        for (int off = 16; off; off >>= 1)
            v = umax64(v, shfl_xor_u64(v, off));
        if (lane == 0) s_wred[wid] = v;
        __syncthreads();
        if (tid < NWAVES) {
            v = s_wred[tid];
#pragma unroll
            for (int off = 16; off; off >>= 1)
                v = umax64(v, shfl_xor_u64(v, off));
            if (tid == 0) s_bcast = v;
        }
        __syncthreads();

        const unsigned best = ~(unsigned)(s_bcast & 0xffffffffull);
        if (tid == 0)
            out[(size_t)b * NSAMP + t] = (float)((unsigned)(b * NPTS) + best);

        // broadcast winner coords from the staged LDS table (same-address broadcast)
        qx = s_pos[3 * best + 0];
        qy = s_pos[3 * best + 1];
        qz = s_pos[3 * best + 2];

        // fused min-update + next-round local argmax, all in VGPRs
        lkey = 0;
#pragma unroll
        for (int k = 0; k < PPT; ++k) {
            const float dx = px[k] - qx, dy = py[k] - qy, dz = pz[k] - qz;
            float d = dx * dx + dy * dy + dz * dz;
            d = fminf(mind[k], d);
            mind[k] = d;
            const unsigned idx = (unsigned)(tid + k * BLOCK);
            lkey = umax64(lkey, ((unsigned long long)__float_as_uint(d) << 32) | (~idx));
        }
    }
}

extern "C" void kernel_launch(void* const* d_in, const int* in_sizes, int n_in,
                              void* d_out, int out_size, void* d_ws, size_t ws_size,
                              hipStream_t stream) {
    const float* pos = (const float*)d_in[0];   // [NCLOUDS*NPTS, 3] fp32
    // d_in[1] (feats) and d_in[2] (batch) are not needed: batch layout is uniform.
    float* out = (float*)d_out;                 // NCLOUDS*NSAMP sample indices
    fps_kernel<<<NCLOUDS, BLOCK, 0, stream>>>(pos, out);
}